// FullyConnected_33509334843513
// MI455X (gfx1250) — compile-verified
//
#include <hip/hip_runtime.h>

typedef __attribute__((ext_vector_type(2))) float v2f;
typedef __attribute__((ext_vector_type(8))) float v8f;

#define IN_DIM 64
#define OUT_DIM 32
#define H1 4
#define TILE_M 16
#define WAVES_PER_BLOCK 8
#define XROW 68   // padded LDS row stride (floats) -> bank-conflict-free fragment loads

__device__ __forceinline__ float fast_sigmoid(float x) {
    // sigmoid(x) = 1 / (1 + exp(-x)) ; exp via v_exp_f32, recip via v_rcp_f32
    float e = __builtin_amdgcn_exp2f(-x * 1.4426950408889634f);
    return __builtin_amdgcn_rcpf(1.0f + e);
}

__global__ __launch_bounds__(256) void mlp4_wmma_kernel(
    const float* __restrict__ x,
    const float* __restrict__ W0, const float* __restrict__ b0,
    const float* __restrict__ W1, const float* __restrict__ b1,
    const float* __restrict__ W2, const float* __restrict__ b2,
    const float* __restrict__ W3, const float* __restrict__ b3,
    float* __restrict__ out, int nTiles)
{
    // LDS layout (floats):
    //  [0,1024)      BW0: W0^T zero-padded to 64x16, packed as k-pairs: BW0[p][n][2] = {B[2p][n], B[2p+1][n]}
    //  [1024,1028)   b0
    //  [1028,1060)   W1 (8x4)
    //  [1060,1068)   b1
    //  [1068,1132)   W2 (8x8)
    //  [1132,1140)   b2
    //  [1140,1396)   W3 (32x8)
    //  [1396,1428)   b3
    //  [1440,10144)  x tiles: 8 waves x 16 rows x 68 (padded) floats
    //  [10144,10656) h0 buffers: 8 waves x 16 rows x 4 floats
    __shared__ float smem[10656];
    float* BW0 = smem;
    float* Lb0 = smem + 1024;
    float* LW1 = smem + 1028;
    float* Lb1 = smem + 1060;
    float* LW2 = smem + 1068;
    float* Lb2 = smem + 1132;
    float* LW3 = smem + 1140;
    float* Lb3 = smem + 1396;
    float* Xt  = smem + 1440;
    float* H0  = smem + 10144;

    const int tid = threadIdx.x;

    // ---- per-block weight preload (uniform across block, before any early return) ----
    for (int i = tid; i < 512; i += 256) {           // 32 k-pairs x 16 cols -> 1024 floats
        int p = i >> 4, n = i & 15;                  // pair p covers k = 2p, 2p+1
        float w0v = 0.f, w1v = 0.f;
        if (n < H1) {                                // zero-pad columns 4..15
            w0v = W0[n * IN_DIM + 2 * p];
            w1v = W0[n * IN_DIM + 2 * p + 1];
        }
        BW0[i * 2 + 0] = w0v;
        BW0[i * 2 + 1] = w1v;
    }
    if (tid < 4)   Lb0[tid] = b0[tid];
    if (tid < 32)  LW1[tid] = W1[tid];
    if (tid < 8)   Lb1[tid] = b1[tid];
    if (tid < 64)  LW2[tid] = W2[tid];
    if (tid < 8)   Lb2[tid] = b2[tid];
    if (tid < 256) LW3[tid] = W3[tid];
    if (tid < 32)  Lb3[tid] = b3[tid];
    __syncthreads();

    const int wave = blockIdx.x * WAVES_PER_BLOCK + (tid >> 5);
    if (wave >= nTiles) return;                      // wave-uniform exit

    const int lane = tid & 31;
    const int half = lane >> 4;                      // 0: lanes 0-15, 1: lanes 16-31
    const int n    = lane & 15;                      // column / row-in-tile index

    float* xw  = Xt + (tid >> 5) * (TILE_M * XROW);
    float* h0w = H0 + (tid >> 5) * (TILE_M * H1);

    // ---- stage 16x64 x-tile into LDS, fully coalesced b128 loads ----
    const float* xg = x + (size_t)wave * (TILE_M * IN_DIM);
    #pragma unroll
    for (int i = 0; i < 8; ++i) {
        int f = i * 128 + lane * 4;                  // flat index within 1024-float tile
        int r = f >> 6, c = f & 63;                  // 4 | 64 so each float4 stays in one row
        float4 v = *(const float4*)(xg + f);
        *(float4*)(xw + r * XROW + c) = v;
    }
    asm volatile("s_wait_dscnt 0" ::: "memory");     // cross-lane LDS visibility (same wave)

    // ---- layer 0: 16 x V_WMMA_F32_16X16X4_F32, bias folded into C ----
    const float bias = (n < H1) ? Lb0[n] : 0.f;
    v8f acc = { bias, bias, bias, bias, bias, bias, bias, bias };
    #pragma unroll
    for (int s = 0; s < 16; ++s) {
        // A fragment: M = lane%16, K = 4s + 2*half (+1)   [ISA 16x4 f32 layout]
        v2f a = *(const v2f*)(xw + n * XROW + 4 * s + 2 * half);
        // B fragment: N = lane%16, K pair index = 2s + half
        v2f b = *(const v2f*)(BW0 + ((2 * s + half) * 16 + n) * 2);
        acc = __builtin_amdgcn_wmma_f32_16x16x4_f32(
            false, a, false, b, (short)0, acc, false, false);
    }

    // ---- sigmoid + scatter valid columns (n<4) back to row-major h0 in LDS ----
    if (n < H1) {
        #pragma unroll
        for (int j = 0; j < 8; ++j) {                // VGPR j -> row 8*half + j
            h0w[(half * 8 + j) * H1 + n] = fast_sigmoid(acc[j]);
        }
    }
    asm volatile("s_wait_dscnt 0" ::: "memory");

    // ---- layers 1-3: per-lane scalar (K<=8, trivially small), weights broadcast from LDS ----
    const int r = n;                                 // each lane owns row lane%16
    float4 h0r = *(const float4*)(h0w + r * H1);

    float h1v[8];
    #pragma unroll
    for (int o = 0; o < 8; ++o) {
        float s = Lb1[o] + h0r.x * LW1[o * 4 + 0] + h0r.y * LW1[o * 4 + 1]
                         + h0r.z * LW1[o * 4 + 2] + h0r.w * LW1[o * 4 + 3];
        h1v[o] = fast_sigmoid(s);
    }
    float h2v[8];
    #pragma unroll
    for (int o = 0; o < 8; ++o) {
        float s = Lb2[o];
        #pragma unroll
        for (int k = 0; k < 8; ++k) s += h1v[k] * LW2[o * 8 + k];
        h2v[o] = fast_sigmoid(s);
    }

    // layer 3: half-wave 0 -> outputs 0..15, half-wave 1 -> outputs 16..31; b128 stores
    float* og = out + ((size_t)wave * TILE_M + r) * OUT_DIM + half * 16;
    #pragma unroll
    for (int q = 0; q < 4; ++q) {
        float res[4];
        #pragma unroll
        for (int t = 0; t < 4; ++t) {
            int o = half * 16 + q * 4 + t;
            float s = Lb3[o];
            #pragma unroll
            for (int k = 0; k < 8; ++k) s += h2v[k] * LW3[o * 8 + k];
            res[t] = fast_sigmoid(s);
        }
        float4 v = { res[0], res[1], res[2], res[3] };
        *(float4*)(og + q * 4) = v;
    }
}

extern "C" void kernel_launch(void* const* d_in, const int* in_sizes, int n_in,
                              void* d_out, int out_size, void* d_ws, size_t ws_size,
                              hipStream_t stream) {
    const float* x  = (const float*)d_in[0];
    const float* W0 = (const float*)d_in[1];
    const float* b0 = (const float*)d_in[2];
    const float* W1 = (const float*)d_in[3];
    const float* b1 = (const float*)d_in[4];
    const float* W2 = (const float*)d_in[5];
    const float* b2 = (const float*)d_in[6];
    const float* W3 = (const float*)d_in[7];
    const float* b3 = (const float*)d_in[8];
    float* out = (float*)d_out;

    int batch  = in_sizes[0] / IN_DIM;               // 1,000,000 (divisible by 16)
    int nTiles = (batch + TILE_M - 1) / TILE_M;      // 62,500
    int blocks = (nTiles + WAVES_PER_BLOCK - 1) / WAVES_PER_BLOCK;

    mlp4_wmma_kernel<<<blocks, 256, 0, stream>>>(x, W0, b0, W1, b1, W2, b2, W3, b3,
                                                 out, nTiles);
}